// _GNNSAGE_62294205662068
// MI455X (gfx1250) — compile-verified
//
#include <hip/hip_runtime.h>
#include <hip/hip_bf16.h>

#define N_USERS  500000
#define N_MOVIES 100000
#define DIMF     64
#define E_EDGES  1250000
#define NEG_SLOPE 0.01f
#define LDSW 72   // padded LDS row stride: 2*72 % 64 == 16 -> lane halves hit disjoint banks

typedef __attribute__((ext_vector_type(2))) float v2f;
typedef __attribute__((ext_vector_type(8))) float v8f;

// ---------------- utility kernels ----------------

__global__ __launch_bounds__(256) void k_zero4(float* __restrict__ p, unsigned n4) {
  unsigned i = blockIdx.x * 256u + threadIdx.x;
  if (i < n4) ((float4*)p)[i] = make_float4(0.f, 0.f, 0.f, 0.f);
}

__global__ __launch_bounds__(256) void k_count(const int* __restrict__ src,
                                               const int* __restrict__ dst,
                                               float* __restrict__ cu,
                                               float* __restrict__ cm, int E) {
  int e = blockIdx.x * 256 + threadIdx.x;
  if (e < E) {
    atomicAdd(&cm[dst[e]], 1.0f);
    atomicAdd(&cu[src[e]], 1.0f);
  }
}

__global__ __launch_bounds__(256) void k_inv(float* __restrict__ c, int n) {
  int i = blockIdx.x * 256 + threadIdx.x;
  if (i < n) c[i] = 1.0f / fmaxf(c[i], 1.0f);
}

// 16 lanes per edge; each lane moves 4 contiguous channels (float4 gather,
// 4x global_atomic_add_f32 scatter; destinations live in L2).
__global__ __launch_bounds__(256) void k_scatter(const float* __restrict__ feat,
                                                 const int* __restrict__ gidx,
                                                 const int* __restrict__ sidx,
                                                 float* __restrict__ agg, int E) {
  unsigned t = blockIdx.x * 256u + threadIdx.x;
  unsigned e = t >> 4;
  if (e >= (unsigned)E) return;
  int c = (int)(t & 15u) << 2;
  int g = gidx[e], s = sidx[e];
  float4 v = *(const float4*)(feat + (size_t)g * DIMF + c);
  float* d = agg + (size_t)s * DIMF + c;
  atomicAdd(d + 0, v.x);
  atomicAdd(d + 1, v.y);
  atomicAdd(d + 2, v.z);
  atomicAdd(d + 3, v.w);
}

// ---------------- fused SAGE transform (WMMA f32 16x16x4) ----------------
// out[r,:] = (agg[r,:]*invc[r]) @ Wl + bias + xs[r,:] @ Wr   (+ optional LeakyReLU)
// One wave -> 16 rows x 64 cols. K=64 in steps of 4 through the f32 WMMA pipe.
__global__ __launch_bounds__(256) void k_sage_gemm(
    const float* __restrict__ agg, const float* __restrict__ invc,
    const float* __restrict__ xs,
    const float* __restrict__ Wl, const float* __restrict__ Wr,
    const float* __restrict__ bias, float* __restrict__ out,
    int nRows, int leaky) {
  __shared__ float sWl[64 * LDSW];
  __shared__ float sWr[64 * LDSW];
  __shared__ float sB[64];

  const int tid = threadIdx.x;
  for (int i = tid; i < 64 * 64; i += 256) {
    int r = i >> 6, c = i & 63;
    sWl[r * LDSW + c] = Wl[i];
    sWr[r * LDSW + c] = Wr[i];
  }
  if (tid < 64) sB[tid] = bias[tid];
  __syncthreads();

  const int wave = tid >> 5;
  const int lane = tid & 31;
  const int tileRow = (blockIdx.x * 8 + wave) * 16;
  if (tileRow >= nRows) return;  // wave-uniform; EXEC stays all-ones for WMMA

  const int l15  = lane & 15;
  const int ksel = (lane >> 4) << 1;        // lanes 0-15 -> K{0,1}; 16-31 -> K{2,3}
  const int mrow = tileRow + l15;
  const float inv = invc[mrow];
  const float* aRow = agg + (size_t)mrow * DIMF + ksel;
  const float* xRow = xs  + (size_t)mrow * DIMF + ksel;

  v8f acc0 = {}, acc1 = {}, acc2 = {}, acc3 = {};

#pragma unroll
  for (int kb = 0; kb < 64; kb += 4) {
    const int kr = kb + ksel;
    v2f a, b;

    // ---- mean-aggregate path: A = agg*inv, B = Wl ----
    {
      float2 t = *(const float2*)(aRow + kb);
      a.x = t.x * inv; a.y = t.y * inv;
    }
    b.x = sWl[(kr + 0) * LDSW +  0 + l15]; b.y = sWl[(kr + 1) * LDSW +  0 + l15];
    acc0 = __builtin_amdgcn_wmma_f32_16x16x4_f32(false, a, false, b, (short)0, acc0, false, false);
    b.x = sWl[(kr + 0) * LDSW + 16 + l15]; b.y = sWl[(kr + 1) * LDSW + 16 + l15];
    acc1 = __builtin_amdgcn_wmma_f32_16x16x4_f32(false, a, false, b, (short)0, acc1, false, false);
    b.x = sWl[(kr + 0) * LDSW + 32 + l15]; b.y = sWl[(kr + 1) * LDSW + 32 + l15];
    acc2 = __builtin_amdgcn_wmma_f32_16x16x4_f32(false, a, false, b, (short)0, acc2, false, false);
    b.x = sWl[(kr + 0) * LDSW + 48 + l15]; b.y = sWl[(kr + 1) * LDSW + 48 + l15];
    acc3 = __builtin_amdgcn_wmma_f32_16x16x4_f32(false, a, false, b, (short)0, acc3, false, false);

    // ---- self path: A = xs, B = Wr ----
    {
      float2 t = *(const float2*)(xRow + kb);
      a.x = t.x; a.y = t.y;
    }
    b.x = sWr[(kr + 0) * LDSW +  0 + l15]; b.y = sWr[(kr + 1) * LDSW +  0 + l15];
    acc0 = __builtin_amdgcn_wmma_f32_16x16x4_f32(false, a, false, b, (short)0, acc0, false, false);
    b.x = sWr[(kr + 0) * LDSW + 16 + l15]; b.y = sWr[(kr + 1) * LDSW + 16 + l15];
    acc1 = __builtin_amdgcn_wmma_f32_16x16x4_f32(false, a, false, b, (short)0, acc1, false, false);
    b.x = sWr[(kr + 0) * LDSW + 32 + l15]; b.y = sWr[(kr + 1) * LDSW + 32 + l15];
    acc2 = __builtin_amdgcn_wmma_f32_16x16x4_f32(false, a, false, b, (short)0, acc2, false, false);
    b.x = sWr[(kr + 0) * LDSW + 48 + l15]; b.y = sWr[(kr + 1) * LDSW + 48 + l15];
    acc3 = __builtin_amdgcn_wmma_f32_16x16x4_f32(false, a, false, b, (short)0, acc3, false, false);
  }

  // D layout: lane<16 -> rows tileRow+0..7, N=lane; lane>=16 -> rows +8..15, N=lane-16
  const int rbase = tileRow + ((lane >> 4) << 3);
#pragma unroll
  for (int m = 0; m < 8; ++m) {
    float* orow = out + (size_t)(rbase + m) * DIMF;
    float v0 = acc0[m] + sB[ 0 + l15];
    float v1 = acc1[m] + sB[16 + l15];
    float v2 = acc2[m] + sB[32 + l15];
    float v3 = acc3[m] + sB[48 + l15];
    if (leaky) {
      v0 = v0 > 0.f ? v0 : v0 * NEG_SLOPE;
      v1 = v1 > 0.f ? v1 : v1 * NEG_SLOPE;
      v2 = v2 > 0.f ? v2 : v2 * NEG_SLOPE;
      v3 = v3 > 0.f ? v3 : v3 * NEG_SLOPE;
    }
    orow[ 0 + l15] = v0;
    orow[16 + l15] = v1;
    orow[32 + l15] = v2;
    orow[48 + l15] = v3;
  }
}

// ---------------- host side ----------------

extern "C" void kernel_launch(void* const* d_in, const int* in_sizes, int n_in,
                              void* d_out, int out_size, void* d_ws, size_t ws_size,
                              hipStream_t stream) {
  const float* x_user  = (const float*)d_in[0];
  const float* x_movie = (const float*)d_in[1];
  const int*   esrc    = (const int*)d_in[2];
  const int*   edst    = (const int*)d_in[3];
  const float *Wl_um0 = (const float*)d_in[4],  *b_um0 = (const float*)d_in[5],  *Wr_um0 = (const float*)d_in[6];
  const float *Wl_mu0 = (const float*)d_in[7],  *b_mu0 = (const float*)d_in[8],  *Wr_mu0 = (const float*)d_in[9];
  const float *Wl_um1 = (const float*)d_in[10], *b_um1 = (const float*)d_in[11], *Wr_um1 = (const float*)d_in[12];
  const float *Wl_mu1 = (const float*)d_in[13], *b_mu1 = (const float*)d_in[14], *Wr_mu1 = (const float*)d_in[15];

  float* out_u = (float*)d_out;
  float* out_m = out_u + (size_t)N_USERS * DIMF;

  char* ws = (char*)d_ws;
  float* agg_u = (float*)(ws + 0);            // 128,000,000 B
  float* agg_m = (float*)(ws + 128000000);    //  25,600,000 B
  float* h_u   = (float*)(ws + 153600000);    // 128,000,000 B
  float* h_m   = (float*)(ws + 281600000);    //  25,600,000 B
  float* cnt_u = (float*)(ws + 307200000);    //   2,000,000 B
  float* cnt_m = (float*)(ws + 309200000);    //     400,000 B

  const int ZB = 256;
  auto nblk = [](size_t n, int b) { return (unsigned)((n + b - 1) / b); };

#define ZERO(p, nfl) k_zero4<<<nblk((nfl) / 4, ZB), ZB, 0, stream>>>((p), (unsigned)((nfl) / 4))

  // degree counts (edge structure shared by both layers)
  ZERO(cnt_u, (size_t)N_USERS);
  ZERO(cnt_m, (size_t)N_MOVIES);
  k_count<<<nblk(E_EDGES, ZB), ZB, 0, stream>>>(esrc, edst, cnt_u, cnt_m, E_EDGES);
  k_inv<<<nblk(N_USERS, ZB), ZB, 0, stream>>>(cnt_u, N_USERS);
  k_inv<<<nblk(N_MOVIES, ZB), ZB, 0, stream>>>(cnt_m, N_MOVIES);

  const unsigned sgrid = nblk((size_t)E_EDGES * 16, ZB);

  // ---- layer 0 ----
  ZERO(agg_u, (size_t)N_USERS * DIMF);
  ZERO(agg_m, (size_t)N_MOVIES * DIMF);
  k_scatter<<<sgrid, ZB, 0, stream>>>(x_user,  esrc, edst, agg_m, E_EDGES);
  k_scatter<<<sgrid, ZB, 0, stream>>>(x_movie, edst, esrc, agg_u, E_EDGES);
  k_sage_gemm<<<nblk(N_MOVIES, 128), ZB, 0, stream>>>(agg_m, cnt_m, x_movie,
      Wl_um0, Wr_um0, b_um0, h_m, N_MOVIES, 1);
  k_sage_gemm<<<nblk(N_USERS, 128), ZB, 0, stream>>>(agg_u, cnt_u, x_user,
      Wl_mu0, Wr_mu0, b_mu0, h_u, N_USERS, 1);

  // ---- layer 1 ----
  ZERO(agg_u, (size_t)N_USERS * DIMF);
  ZERO(agg_m, (size_t)N_MOVIES * DIMF);
  k_scatter<<<sgrid, ZB, 0, stream>>>(h_u, esrc, edst, agg_m, E_EDGES);
  k_scatter<<<sgrid, ZB, 0, stream>>>(h_m, edst, esrc, agg_u, E_EDGES);
  k_sage_gemm<<<nblk(N_MOVIES, 128), ZB, 0, stream>>>(agg_m, cnt_m, h_m,
      Wl_um1, Wr_um1, b_um1, out_m, N_MOVIES, 0);
  k_sage_gemm<<<nblk(N_USERS, 128), ZB, 0, stream>>>(agg_u, cnt_u, h_u,
      Wl_mu1, Wr_mu1, b_mu1, out_u, N_USERS, 0);

#undef ZERO
}